// GNNModel_39917426049337
// MI455X (gfx1250) — compile-verified
//
#include <hip/hip_runtime.h>

typedef __attribute__((ext_vector_type(2))) float v2f;
typedef __attribute__((ext_vector_type(8))) float v8f;

#define F48 48

// ---------------------------------------------------------------------------
// Kernel 1: feat = concat(x, h, q) per node; also zero the degree counter.
// ---------------------------------------------------------------------------
__global__ void build_feat_zero_deg(const float* __restrict__ x,
                                    const float* __restrict__ h,
                                    const float* __restrict__ q,
                                    float* __restrict__ featw,
                                    int* __restrict__ deg, int N) {
  int i = blockIdx.x * blockDim.x + threadIdx.x;
  if (i >= N) return;
  const float4* xr = (const float4*)(x + (size_t)i * 16);
  const float4* hr = (const float4*)(h + (size_t)i * 16);
  const float4* qr = (const float4*)(q + (size_t)i * 16);
  float4* fw = (float4*)(featw + (size_t)i * F48);
#pragma unroll
  for (int k = 0; k < 4; ++k) fw[k] = xr[k];
#pragma unroll
  for (int k = 0; k < 4; ++k) fw[4 + k] = hr[k];
#pragma unroll
  for (int k = 0; k < 4; ++k) fw[8 + k] = qr[k];
  deg[i] = 0;
}

// ---------------------------------------------------------------------------
// Kernel 2: indegree count (matches segment_sum over dst).
// ---------------------------------------------------------------------------
__global__ void count_deg(const int* __restrict__ dst, int* __restrict__ deg,
                          int E) {
  int i = blockIdx.x * blockDim.x + threadIdx.x;
  if (i >= E) return;
  atomicAdd(&deg[dst[i]], 1);
}

// ---------------------------------------------------------------------------
// Kernel 3: dis = rsqrt(deg+1), dis2 = 1/(deg+1).
// ---------------------------------------------------------------------------
__global__ void compute_dis(const int* __restrict__ deg,
                            float* __restrict__ dis, float* __restrict__ dis2,
                            int N) {
  int i = blockIdx.x * blockDim.x + threadIdx.x;
  if (i >= N) return;
  float d = (float)(deg[i] + 1);
  dis[i] = rsqrtf(d);
  dis2[i] = 1.0f / d;
}

// ---------------------------------------------------------------------------
// Kernel 4: xw = feat @ W3^T via V_WMMA_F32_16X16X4_F32.
// Each wave owns a 16-row block and the full 48 output columns (3 tiles).
// A layout (16x4 f32): lane l<16 holds row l, {K0,K1}; lane l+16 holds {K2,K3}.
// B layout (4x16 f32): mirrored; lane = column, half-wave selects K pair.
// C/D layout: vgpr v, lanes 0-15 -> (M=v, N=lane); lanes 16-31 -> (M=v+8).
// Epilogue fuses self-loop (xw * dis2) + bias into `out`, and stores xw.
// ---------------------------------------------------------------------------
__global__ void gcn_gemm_wmma(const float* __restrict__ featw,
                              const float* __restrict__ W3,  // [48 out][48 in]
                              const float* __restrict__ b3,  // [48]
                              const float* __restrict__ dis2,
                              float* __restrict__ xw,
                              float* __restrict__ out, int N) {
  const int lane = threadIdx.x & 31;
  const int wave = threadIdx.x >> 5;
  const int nRowBlocks = (N + 15) >> 4;
  const int rowBlock = blockIdx.x * (blockDim.x >> 5) + wave;
  if (rowBlock >= nRowBlocks) return;  // wave-uniform: EXEC all-ones for WMMA
  const int row0 = rowBlock << 4;

  const int lm = lane & 15;             // row (A) / column (B,C,D) within tile
  const int khalf = (lane >> 4) << 1;   // K sub-pair: 0 or 2

  int arow = row0 + lm;
  if (arow >= N) arow = N - 1;  // clamped read (N%16==0 here, but be safe)

  v8f acc0 = {};
  v8f acc1 = {};
  v8f acc2 = {};
#pragma unroll
  for (int k = 0; k < 12; ++k) {
    const int k4 = (k << 2) + khalf;
    v2f a  = *(const v2f*)(featw + (size_t)arow * F48 + k4);
    v2f b0 = *(const v2f*)(W3 + (size_t)(0  + lm) * F48 + k4);
    v2f b1 = *(const v2f*)(W3 + (size_t)(16 + lm) * F48 + k4);
    v2f b2 = *(const v2f*)(W3 + (size_t)(32 + lm) * F48 + k4);
    acc0 = __builtin_amdgcn_wmma_f32_16x16x4_f32(false, a, false, b0, (short)0,
                                                 acc0, false, false);
    acc1 = __builtin_amdgcn_wmma_f32_16x16x4_f32(false, a, false, b1, (short)0,
                                                 acc1, false, false);
    acc2 = __builtin_amdgcn_wmma_f32_16x16x4_f32(false, a, false, b2, (short)0,
                                                 acc2, false, false);
  }

  const int mbase = (lane >> 4) << 3;  // 0 or 8
  const float bc0 = b3[lm];
  const float bc1 = b3[16 + lm];
  const float bc2 = b3[32 + lm];
#pragma unroll
  for (int v = 0; v < 8; ++v) {
    int m = row0 + mbase + v;
    if (m < N) {
      float d2 = dis2[m];
      float e0 = acc0[v], e1 = acc1[v], e2 = acc2[v];
      size_t base = (size_t)m * F48;
      xw[base + lm]      = e0;
      xw[base + 16 + lm] = e1;
      xw[base + 32 + lm] = e2;
      out[base + lm]      = e0 * d2 + bc0;
      out[base + 16 + lm] = e1 * d2 + bc1;
      out[base + 32 + lm] = e2 * d2 + bc2;
    }
  }
}

// ---------------------------------------------------------------------------
// Kernel 5: edge scatter: out[dst] += xw[src] * dis[src]*dis[dst].
// xw (19.2MB) is L2-resident on MI455X (192MB L2); fp32 atomics stay in L2.
// ---------------------------------------------------------------------------
__global__ void scatter_edges(const int* __restrict__ src,
                              const int* __restrict__ dst,
                              const float* __restrict__ dis,
                              const float* __restrict__ xw,
                              float* __restrict__ out, int E) {
  int i = blockIdx.x * blockDim.x + threadIdx.x;
  if (i >= E) return;
  int s = src[i];
  int t = dst[i];
  float nm = dis[s] * dis[t];
  const float4* xr = (const float4*)(xw + (size_t)s * F48);
  float* orow = out + (size_t)t * F48;
#pragma unroll
  for (int v = 0; v < 12; ++v) {
    float4 val = xr[v];
    __hip_atomic_fetch_add(orow + v * 4 + 0, val.x * nm, __ATOMIC_RELAXED,
                           __HIP_MEMORY_SCOPE_AGENT);
    __hip_atomic_fetch_add(orow + v * 4 + 1, val.y * nm, __ATOMIC_RELAXED,
                           __HIP_MEMORY_SCOPE_AGENT);
    __hip_atomic_fetch_add(orow + v * 4 + 2, val.z * nm, __ATOMIC_RELAXED,
                           __HIP_MEMORY_SCOPE_AGENT);
    __hip_atomic_fetch_add(orow + v * 4 + 3, val.w * nm, __ATOMIC_RELAXED,
                           __HIP_MEMORY_SCOPE_AGENT);
  }
}

// ---------------------------------------------------------------------------
// Host launcher. Input order (setup_inputs dict): h, e, x, q, mask, W, b.
// Only the last layer's W/b matter (reference overwrites `out` each t).
// ---------------------------------------------------------------------------
extern "C" void kernel_launch(void* const* d_in, const int* in_sizes, int n_in,
                              void* d_out, int out_size, void* d_ws,
                              size_t ws_size, hipStream_t stream) {
  const float* h = (const float*)d_in[0];
  const int* e = (const int*)d_in[1];
  const float* x = (const float*)d_in[2];
  const float* q = (const float*)d_in[3];
  // d_in[4] = mask (unused by the reference computation)
  const float* W = (const float*)d_in[5];
  const float* b = (const float*)d_in[6];

  const int N = in_sizes[0] / 16;
  const int E = in_sizes[1] / 2;
  const int T = in_sizes[6] / F48;  // b is [T, 48]

  const float* W3 = W + (size_t)(T - 1) * F48 * F48;
  const float* b3 = b + (size_t)(T - 1) * F48;
  const int* srcp = e;      // e[0, :]
  const int* dstp = e + E;  // e[1, :]
  float* out = (float*)d_out;

  // Workspace layout: featw[N*48] | xw[N*48] | dis[N] | dis2[N] | deg[N]
  float* featw = (float*)d_ws;
  float* xw = featw + (size_t)N * F48;
  float* dis = xw + (size_t)N * F48;
  float* dis2 = dis + N;
  int* deg = (int*)(dis2 + N);

  const int BLK = 256;
  build_feat_zero_deg<<<(N + BLK - 1) / BLK, BLK, 0, stream>>>(x, h, q, featw,
                                                               deg, N);
  count_deg<<<(E + BLK - 1) / BLK, BLK, 0, stream>>>(dstp, deg, E);
  compute_dis<<<(N + BLK - 1) / BLK, BLK, 0, stream>>>(deg, dis, dis2, N);

  const int nRowBlocks = (N + 15) / 16;   // 6250 for N=100000
  const int wavesPerBlock = 8;            // 256 threads, wave32
  gcn_gemm_wmma<<<(nRowBlocks + wavesPerBlock - 1) / wavesPerBlock, BLK, 0,
                  stream>>>(featw, W3, b3, dis2, xw, out, N);

  scatter_edges<<<(E + BLK - 1) / BLK, BLK, 0, stream>>>(srcp, dstp, dis, xw,
                                                         out, E);
}